// UNetInnerProd_18210661335355
// MI455X (gfx1250) — compile-verified
//
#include <hip/hip_runtime.h>

// Geometry fixed by the reference
#define H_  72
#define W_  96
#define N_  6912
#define B_  2
#define C_  3
#define NT  432           // N_/16 tiles per dimension
#define JCHUNK 16         // j-tiles handled per wave
#define NCHUNK (NT / JCHUNK)  // 27 strips per row of tiles

typedef __attribute__((ext_vector_type(2))) float v2f;
typedef __attribute__((ext_vector_type(8))) float v8f;

// -0.1 * log2(e): exp(-0.1*x) == exp2(C*x)
#define NEG01_LOG2E (-0.14426950408889634f)

__global__ void zero_out_kernel(float* __restrict__ o) {
    if (threadIdx.x < 3) o[threadIdx.x] = 0.0f;
}

// One thread per (b, n) point: build WMMA operand tables in workspace.
//  A1[b][n] = (-2*x1, -2*y1, -2*z1, 1)            (A operand, distance-folded)
//  Bc[b][n] = (x2c, y2c, z2c, |x2c|^2)            (B operand, clean transform)
//  Bn[b][n] = (x2n, y2n, z2n, |x2n|^2)            (B operand, noisy transform)
//  G1[b][n] = f1/||f1|| padded with 0             (A operand for gram)
//  G2[b][n] = f2/||f2|| padded with 0             (B operand for gram)
//  S1[b][n] = |x1|^2
// Also accumulates 100*(||f1|| + ||f2||) into out[2].
__global__ void stage1_prep(const float* __restrict__ f1,
                            const float* __restrict__ f2,
                            const float* __restrict__ d1,
                            const float* __restrict__ d2,
                            const float* __restrict__ pose,
                            const float* __restrict__ noise,
                            const float* __restrict__ grid,
                            float* __restrict__ A1,
                            float* __restrict__ Bc,
                            float* __restrict__ Bn,
                            float* __restrict__ G1,
                            float* __restrict__ G2,
                            float* __restrict__ S1,
                            float* __restrict__ out)
{
    const int t = blockIdx.x * blockDim.x + threadIdx.x;
    float contrib = 0.0f;
    if (t < B_ * N_) {
        const int b = t / N_;
        const int n = t % N_;

        const float g0 = grid[n];
        const float g1 = grid[N_ + n];
        const float g2 = grid[2 * N_ + n];

        // --- camera-1 points ---
        const float dd1 = d1[t];
        const float x1 = g0 * dd1, y1 = g1 * dd1, z1 = g2 * dd1;
        A1[4 * t + 0] = -2.0f * x1;
        A1[4 * t + 1] = -2.0f * y1;
        A1[4 * t + 2] = -2.0f * z1;
        A1[4 * t + 3] = 1.0f;
        S1[t] = x1 * x1 + y1 * y1 + z1 * z1;

        // --- camera-2 points, transformed ---
        const float dd2 = d2[t];
        const float x2 = g0 * dd2, y2 = g1 * dd2, z2 = g2 * dd2;
        const float* P = pose + b * 16;

        const float xc = P[0] * x2 + P[1] * y2 + P[2]  * z2 + P[3];
        const float yc = P[4] * x2 + P[5] * y2 + P[6]  * z2 + P[7];
        const float zc = P[8] * x2 + P[9] * y2 + P[10] * z2 + P[11];
        Bc[4 * t + 0] = xc;
        Bc[4 * t + 1] = yc;
        Bc[4 * t + 2] = zc;
        Bc[4 * t + 3] = xc * xc + yc * yc + zc * zc;

        // noisy pose rows: PN[r][c] = sum_k P[r][k] * noise[k][c]
        float pn[12];
        #pragma unroll
        for (int r = 0; r < 3; ++r) {
            #pragma unroll
            for (int c = 0; c < 4; ++c) {
                pn[r * 4 + c] = P[r * 4 + 0] * noise[0 * 4 + c]
                              + P[r * 4 + 1] * noise[1 * 4 + c]
                              + P[r * 4 + 2] * noise[2 * 4 + c]
                              + P[r * 4 + 3] * noise[3 * 4 + c];
            }
        }
        const float xn = pn[0] * x2 + pn[1] * y2 + pn[2]  * z2 + pn[3];
        const float yn = pn[4] * x2 + pn[5] * y2 + pn[6]  * z2 + pn[7];
        const float zn = pn[8] * x2 + pn[9] * y2 + pn[10] * z2 + pn[11];
        Bn[4 * t + 0] = xn;
        Bn[4 * t + 1] = yn;
        Bn[4 * t + 2] = zn;
        Bn[4 * t + 3] = xn * xn + yn * yn + zn * zn;

        // --- normalized features ---
        const float a0 = f1[b * C_ * N_ + 0 * N_ + n];
        const float a1 = f1[b * C_ * N_ + 1 * N_ + n];
        const float a2 = f1[b * C_ * N_ + 2 * N_ + n];
        const float nrm1 = __builtin_amdgcn_sqrtf(a0 * a0 + a1 * a1 + a2 * a2);
        const float i1 = 1.0f / nrm1;
        G1[4 * t + 0] = a0 * i1;
        G1[4 * t + 1] = a1 * i1;
        G1[4 * t + 2] = a2 * i1;
        G1[4 * t + 3] = 0.0f;

        const float b0 = f2[b * C_ * N_ + 0 * N_ + n];
        const float b1 = f2[b * C_ * N_ + 1 * N_ + n];
        const float b2 = f2[b * C_ * N_ + 2 * N_ + n];
        const float nrm2 = __builtin_amdgcn_sqrtf(b0 * b0 + b1 * b1 + b2 * b2);
        const float i2 = 1.0f / nrm2;
        G2[4 * t + 0] = b0 * i2;
        G2[4 * t + 1] = b1 * i2;
        G2[4 * t + 2] = b2 * i2;
        G2[4 * t + 3] = 0.0f;

        contrib = 100.0f * (nrm1 + nrm2);
    }
    // wave32 reduction, then one atomic per wave
    #pragma unroll
    for (int off = 16; off > 0; off >>= 1)
        contrib += __shfl_xor(contrib, off, 32);
    if ((threadIdx.x & 31) == 0)
        atomicAdd(out + 2, contrib);
}

// Load the float2 operand slice for this lane: lanes 0-15 take components
// {0,1} of row (base + lane&15); lanes 16-31 take components {2,3}.
// This matches both the 16x4 A layout and the 4x16 B layout of
// V_WMMA_F32_16X16X4_F32 (wave32).
__device__ __forceinline__ v2f load_op(const float* __restrict__ tab,
                                       int base, int l15, int half) {
    return *(const v2f*)(tab + (size_t)(base + l15) * 4 + 2 * half);
}

// One wave per (batch, i-tile, j-strip). Each wave computes a 16-row by
// 256-col (16 j-tiles) band of the pairwise reduction.
__global__ void stage2_tiles(const float* __restrict__ A1,
                             const float* __restrict__ Bc,
                             const float* __restrict__ Bn,
                             const float* __restrict__ G1,
                             const float* __restrict__ G2,
                             const float* __restrict__ S1,
                             float* __restrict__ out)
{
    const int gtid = blockIdx.x * blockDim.x + threadIdx.x;
    const int wave = gtid >> 5;
    const int lane = threadIdx.x & 31;
    const int l15  = lane & 15;
    const int half = lane >> 4;

    const int wavesTotal = B_ * NT * NCHUNK;  // 23328
    if (wave >= wavesTotal) return;

    const int b   = wave / (NT * NCHUNK);
    const int rem = wave % (NT * NCHUNK);
    const int ti  = rem / NCHUNK;
    const int jc  = rem % NCHUNK;

    const int i0    = b * N_ + ti * 16;
    const int jbase = b * N_ + jc * (JCHUNK * 16);

    // A operands: loop-invariant across the j-strip
    const v2f a1 = load_op(A1, i0, l15, half);
    const v2f ag = load_op(G1, i0, l15, half);

    // s1 for the 8 rows this lane's accumulator entries cover (loop-invariant)
    float s1v[8];
    #pragma unroll
    for (int e = 0; e < 8; ++e)
        s1v[e] = S1[i0 + e + 8 * half];

    float sum = 0.0f;
    for (int jt = 0; jt < JCHUNK; ++jt) {
        const int j0 = jbase + jt * 16;
        const v2f bc = load_op(Bc, j0, l15, half);
        const v2f bn = load_op(Bn, j0, l15, half);
        const v2f bg = load_op(G2, j0, l15, half);

        v8f zero = {};
        // crossC[e] = s2c_j - 2*dot(x1_i, x2c_j)  (bias folded into operands)
        v8f crossC = __builtin_amdgcn_wmma_f32_16x16x4_f32(
            false, a1, false, bc, (short)0, zero, false, false);
        v8f crossN = __builtin_amdgcn_wmma_f32_16x16x4_f32(
            false, a1, false, bn, (short)0, zero, false, false);
        v8f gram = __builtin_amdgcn_wmma_f32_16x16x4_f32(
            false, ag, false, bg, (short)0, zero, false, false);

        #pragma unroll
        for (int e = 0; e < 8; ++e) {
            const float d2c = fmaxf(crossC[e] + s1v[e], 0.0f);
            const float d2n = fmaxf(crossN[e] + s1v[e], 0.0f);
            // exp(-0.1*sqrt(d2)) == exp2(NEG01_LOG2E * sqrt(d2)); native
            // v_sqrt_f32 / v_exp_f32 (TRANS32, ~1ulp) instead of the IEEE
            // library expansions.
            const float kc = __builtin_amdgcn_exp2f(
                NEG01_LOG2E * __builtin_amdgcn_sqrtf(d2c));
            const float kn = __builtin_amdgcn_exp2f(
                NEG01_LOG2E * __builtin_amdgcn_sqrtf(d2n));
            // inner_neg = sum (k_noisy - k_clean) * gram
            sum += (kn - kc) * gram[e];
        }
    }

    // wave32 reduction, one atomic per wave
    #pragma unroll
    for (int off = 16; off > 0; off >>= 1)
        sum += __shfl_xor(sum, off, 32);
    if (lane == 0) {
        atomicAdd(out + 0, sum);  // final_loss
        atomicAdd(out + 1, sum);  // inner_neg (identical)
    }
}

extern "C" void kernel_launch(void* const* d_in, const int* in_sizes, int n_in,
                              void* d_out, int out_size, void* d_ws, size_t ws_size,
                              hipStream_t stream) {
    const float* feature1 = (const float*)d_in[0];
    const float* feature2 = (const float*)d_in[1];
    const float* depth1   = (const float*)d_in[2];
    const float* depth2   = (const float*)d_in[3];
    const float* pose     = (const float*)d_in[4];
    const float* noise    = (const float*)d_in[5];
    const float* grid     = (const float*)d_in[6];

    float* out = (float*)d_out;
    float* ws  = (float*)d_ws;

    const int BN4 = B_ * N_ * 4;   // 55296 floats per table
    float* A1 = ws;
    float* Bc = A1 + BN4;
    float* Bn = Bc + BN4;
    float* G1 = Bn + BN4;
    float* G2 = G1 + BN4;
    float* S1 = G2 + BN4;          // B_*N_ floats; total ~1.16 MB

    zero_out_kernel<<<1, 64, 0, stream>>>(out);

    const int pts = B_ * N_;       // 13824
    stage1_prep<<<(pts + 255) / 256, 256, 0, stream>>>(
        feature1, feature2, depth1, depth2, pose, noise, grid,
        A1, Bc, Bn, G1, G2, S1, out);

    const int wavesTotal = B_ * NT * NCHUNK;          // 23328 waves
    const int blocks = (wavesTotal * 32 + 255) / 256; // 2916 blocks of 8 waves
    stage2_tiles<<<blocks, 256, 0, stream>>>(A1, Bc, Bn, G1, G2, S1, out);
}